// FuzzyLogic2_57501022158956
// MI455X (gfx1250) — compile-verified
//
#include <hip/hip_runtime.h>
#include <math.h>

// ---------------------------------------------------------------------------
// FuzzyLogic2 on MI455X (gfx1250, wave32)
//
// Math: and(x,y)=xy/(x+y-xy) with p=0 satisfies 1/and - 1 = (1/x-1)+(1/y-1).
// So with u(v) = (1-v)/v the sequential 119-step chain collapses to
//   result = 1 / (1 + sum_{k=0..119} u(maze[row, col_k]))
//   -log(clamp(result, e^-90)) = min(log1p(sum), 90)
// Output = mean over 16384 rows.
//
// Implementation: one wave handles 16 rows. The 120-element row sums are
// computed as S = U * ones via chained V_WMMA_F32_16X16X4_F32 (30 K-chunks),
// which puts the reduction on the CDNA5 matrix pipe. Each lane issues exactly
// two 4-byte gather loads per chunk (every needed element read once ->
// ~150 MB of HBM traffic, ~6-7 us at 23.3 TB/s; compute is negligible).
// Deterministic two-stage reduction (no float atomics).
// ---------------------------------------------------------------------------

typedef __attribute__((ext_vector_type(2))) float v2f;
typedef __attribute__((ext_vector_type(8))) float v8f;

#define BATCH            16384
#define MAZE_LEN         3844
#define NCHUNK           30     // 120 gathered columns / K=4 per WMMA
#define ROWS_PER_WAVE    16
#define WAVES_PER_BLOCK  8
#define THREADS          256
#define NBLOCKS          (BATCH / (ROWS_PER_WAVE * WAVES_PER_BLOCK))  // 128

// Column index for flattened gather position k (0..119):
//  k=0 -> IDX_A=6, k=1 -> IDX_B=3839
//  k=2..63   : i=(k-2)/2 ;  even: 124*i+1 , odd: 124*i+121   (31 pairs)
//  k=64..119 : i=(k-64)/2;  even: 4*i+9   , odd: 4*i+3721    (28 pairs)
__host__ __device__ constexpr int colAt(int k) {
    if (k == 0) return 6;
    if (k == 1) return 3839;
    if (k < 64) { int i = (k - 2) >> 1; return (k & 1) ? (124 * i + 121) : (124 * i + 1); }
    int i = (k - 64) >> 1;              return (k & 1) ? (4 * i + 3721)  : (4 * i + 9);
}

__global__ __launch_bounds__(THREADS)
void fuzzy_main(const float* __restrict__ maze, float* __restrict__ partial) {
    const int tid  = threadIdx.x;
    const int lane = tid & 31;
    const int wave = tid >> 5;
    const int h    = lane >> 4;    // 0: this lane supplies K=0,1 ; 1: K=2,3
    const int m    = lane & 15;    // matrix row M within the 16-row tile

    const int row = (blockIdx.x * WAVES_PER_BLOCK + wave) * ROWS_PER_WAVE + m;
    const float* __restrict__ rowp = maze + (size_t)row * MAZE_LEN;

    v8f acc = {};                       // C/D accumulator (16x16 f32, col 0..15 identical)
    v2f bones;                          // B = ones(4x16): D[r][n] = sum_k A[r][k]
    bones[0] = 1.0f;
    bones[1] = 1.0f;

#pragma unroll
    for (int ch = 0; ch < NCHUNK; ++ch) {
        // Compile-time column constants for this chunk; select pair by lane half.
        const int a0 = colAt(4 * ch + 0), a1 = colAt(4 * ch + 1);
        const int b0 = colAt(4 * ch + 2), b1 = colAt(4 * ch + 3);
        const int c0 = h ? b0 : a0;
        const int c1 = h ? b1 : a1;

        const float x0 = rowp[c0];
        const float x1 = rowp[c1];

        v2f a;                          // A-matrix layout: VGPR0 = K(+0), VGPR1 = K(+1)
        a[0] = (1.0f - x0) / x0;        // u = 1/v - 1  (>= 0 for v in (0,1])
        a[1] = (1.0f - x1) / x1;

        // D = A(16x4,f32) * B(4x16,ones) + C  -> accumulates row sums
        acc = __builtin_amdgcn_wmma_f32_16x16x4_f32(
            /*neg_a=*/false, a, /*neg_b=*/false, bones,
            /*c_mod=*/(short)0, acc, /*reuse_a=*/false, /*reuse_b=*/false);
    }

    // With ones-B every lane holds the same per-row sums:
    // lanes 0-15: acc[j] = S[row j] ; lanes 16-31: acc[j] = S[row 8+j]
    float tsum = 0.0f;
#pragma unroll
    for (int j = 0; j < 8; ++j) {
        float t = log1pf(acc[j]);       // == -log(result); inf -> clamped below
        tsum += fminf(t, 90.0f);        // clamp(result, e^-90)
    }

    // Deterministic block reduction: 2 contributors per wave (lane 0 and 16).
    __shared__ float red[WAVES_PER_BLOCK * 2];
    if (m == 0) red[wave * 2 + h] = tsum;
    __syncthreads();
    if (tid == 0) {
        float s = 0.0f;
#pragma unroll
        for (int i = 0; i < WAVES_PER_BLOCK * 2; ++i) s += red[i];
        partial[blockIdx.x] = s;
    }
}

__global__ void fuzzy_final(const float* __restrict__ partial, float* __restrict__ out) {
    if (threadIdx.x == 0 && blockIdx.x == 0) {
        float s = 0.0f;
        for (int i = 0; i < NBLOCKS; ++i) s += partial[i];   // fixed order -> deterministic
        out[0] = s * (1.0f / (float)BATCH);
    }
}

extern "C" void kernel_launch(void* const* d_in, const int* in_sizes, int n_in,
                              void* d_out, int out_size, void* d_ws, size_t ws_size,
                              hipStream_t stream) {
    const float* maze    = (const float*)d_in[0];   // (16384, 3844) float32
    float*       out     = (float*)d_out;           // scalar float32
    float*       partial = (float*)d_ws;            // 128 floats of scratch

    fuzzy_main<<<NBLOCKS, THREADS, 0, stream>>>(maze, partial);
    fuzzy_final<<<1, 32, 0, stream>>>(partial, out);
}